// ConvCaps_9380208575185
// MI455X (gfx1250) — compile-verified
//
#include <hip/hip_runtime.h>
#include <math.h>

typedef __attribute__((ext_vector_type(2))) float v2f;
typedef __attribute__((ext_vector_type(8))) float v8f;

#define KKB   288          // K*K*B_IN
#define CO    32           // C_OUT
#define PS    16           // P*P
#define LPB   4            // locations (output sites) per block
#define NLOC  576          // 16 * 6 * 6
#define EPSF  1e-8f
#define LAM   1e-3f
#define LN2PI 1.8378770664093453f

// LDS layout in floats
#define POSE_OFF 0
#define A_OFF    (POSE_OFF + LPB*KKB*PS)      // 18432 floats
#define R_OFF    (A_OFF + LPB*KKB)            // +1152
#define RSUM_OFF (R_OFF + LPB*KKB*CO)         // +36864
#define SMEMF    (RSUM_OFF + LPB*CO)          // +128  -> 56576 floats = 226304 B

__device__ __forceinline__ float qsum4(float v) {
    // sum across the 4 adjacent lanes sharing one output capsule c (l = 0..3)
    v += __shfl_xor(v, 1, 32);
    v += __shfl_xor(v, 2, 32);
    return v;
}

// One 16x16 vote tile for input capsule n:
//   rows  = (loc_local, i), cols = (c_local, l), K = j (4)
__device__ __forceinline__ v8f vote_tile(const float* __restrict__ Wg,
                                         const float* poseS,
                                         int n, int lane, int wv) {
    const int m   = lane & 15;            // A row / B column
    const int j0  = (lane >> 4) << 1;     // lanes 0-15: K=0,1 ; lanes 16-31: K=2,3
    const int loc = m >> 2;
    const int i   = m & 3;
    v2f A, B;
    const float* pp = poseS + ((loc * KKB + n) << 4) + (i << 2) + j0;
    A.x = pp[0];
    A.y = pp[1];
    const int c = (wv << 2) + (m >> 2);   // this wave's c slab
    const int l = m & 3;
    const float* wp = Wg + ((((n * CO + c) << 2) + j0) << 2) + l;
    B.x = wp[0];
    B.y = wp[4];                           // j0+1 is 4 floats away
    v8f z = {0.f, 0.f, 0.f, 0.f, 0.f, 0.f, 0.f, 0.f};
    return __builtin_amdgcn_wmma_f32_16x16x4_f32(false, A, false, B,
                                                 (short)0, z, false, false);
}

__global__ void __launch_bounds__(256, 1)
convcaps_em_kernel(const float* __restrict__ x, const float* __restrict__ a,
                   const float* __restrict__ Wg, const float* __restrict__ beta_u,
                   const float* __restrict__ beta_a, float* __restrict__ out) {
    extern __shared__ float smem[];
    float* poseS = smem + POSE_OFF;
    float* aS    = smem + A_OFF;
    float* rS    = smem + R_OFF;
    float* rsumS = smem + RSUM_OFF;

    const int tid  = threadIdx.x;
    const int wv   = tid >> 5;
    const int lane = tid & 31;
    const int locBase = blockIdx.x * LPB;

    // ---- stage pose patches into LDS (float4-wide: b128 in, ds_store_b128 out) ----
    for (int idx = tid; idx < LPB * KKB * 4; idx += 256) {
        const int locl = idx / (KKB * 4);
        const int rem  = idx - locl * (KKB * 4);
        const int n  = rem >> 2;
        const int e4 = rem & 3;
        const int kh = n / 96, kw = (n % 96) >> 5, bi = n & 31;
        const int loc = locBase + locl;
        const int bb = loc / 36, r2 = loc - bb * 36;
        const int oh = r2 / 6, ow = r2 - oh * 6;
        const int ih = oh * 2 + kh, iw = ow * 2 + kw;
        const float4* src = (const float4*)(x + ((((bb * 14 + ih) * 14 + iw) * 32 + bi) << 4));
        ((float4*)poseS)[idx] = src[e4];
    }
    // ---- stage activation patches into LDS ----
    for (int idx = tid; idx < LPB * KKB; idx += 256) {
        const int locl = idx / KKB;
        const int n    = idx - locl * KKB;
        const int kh = n / 96, kw = (n % 96) >> 5, bi = n & 31;
        const int loc = locBase + locl;
        const int bb = loc / 36, r2 = loc - bb * 36;
        const int oh = r2 / 6, ow = r2 - oh * 6;
        const int ih = oh * 2 + kh, iw = ow * 2 + kw;
        aS[idx] = a[((bb * 14 + ih) * 14 + iw) * 32 + bi];
    }

    // per-lane EM mapping (matches f32 16x16 D tile layout)
    const int cc   = (wv << 2) + ((lane & 15) >> 2); // output capsule c
    const int l    = lane & 3;                       // pose column l
    const int half = lane >> 4;
    const int locA = half * 2;                       // VGPRs 0-3 rows
    const int locB = half * 2 + 1;                   // VGPRs 4-7 rows
    const float bu = beta_u[cc];
    const float ba = beta_a[cc];

    float mu[8], sg[8];
    float logaA = 0.f, logaB = 0.f;
#pragma unroll
    for (int g = 0; g < 8; ++g) { mu[g] = 0.f; sg[g] = 1.f; }

    for (int it = 0; it < 3; ++it) {
        __syncthreads();
        if (it == 0) {
            // uniform routing: r = a_in / C
            for (int idx = tid; idx < LPB * KKB * CO; idx += 256) {
                const int n    = (idx >> 5) % KKB;
                const int locl = idx / (KKB * CO);
                rS[idx] = aS[locl * KKB + n] * (1.f / 32.f);
            }
        } else {
            // hoist 1/(2*sigma) and 0.5*log(sigma)+0.5*ln(2pi): loop-invariant over n
            float inv2s[8], cstA = 0.f, cstB = 0.f;
#pragma unroll
            for (int g = 0; g < 8; ++g) {
                inv2s[g] = 0.5f * __builtin_amdgcn_rcpf(sg[g]);
                const float t = 0.5f * logf(sg[g]) + 0.5f * LN2PI;
                if (g < 4) cstA += t; else cstB += t;
            }
            // ---- pass A: ln_ap[loc][n][c] from recomputed votes ----
            for (int n = 0; n < KKB; ++n) {
                __builtin_prefetch(Wg + (n + 8) * CO * PS, 0, 1);
                v8f v = vote_tile(Wg, poseS, n, lane, wv);
                float sA = -cstA, sB = -cstB;
#pragma unroll
                for (int g = 0; g < 4; ++g) {
                    const float d = v[g] - mu[g];
                    sA -= d * d * inv2s[g];
                }
#pragma unroll
                for (int g = 4; g < 8; ++g) {
                    const float d = v[g] - mu[g];
                    sB -= d * d * inv2s[g];
                }
                sA = qsum4(sA) + logaA;
                sB = qsum4(sB) + logaB;
                if (l == 0) {
                    rS[((locA * KKB + n) << 5) + cc] = sA;
                    rS[((locB * KKB + n) << 5) + cc] = sB;
                }
            }
            __syncthreads();
            // ---- softmax over c per (loc, n) row ----
            for (int row = tid; row < LPB * KKB; row += 256) {
                float* p = rS + (row << 5);
                float mx = p[0];
#pragma unroll
                for (int c = 1; c < CO; ++c) mx = fmaxf(mx, p[c]);
                float s = 0.f;
#pragma unroll
                for (int c = 0; c < CO; ++c) { const float e = expf(p[c] - mx); p[c] = e; s += e; }
                const float inv = __builtin_amdgcn_rcpf(s);
#pragma unroll
                for (int c = 0; c < CO; ++c) p[c] *= inv;
            }
        }
        __syncthreads();
        // ---- r_sum over n ----
        if (tid < LPB * CO) {
            const int locl = tid >> 5, c = tid & 31;
            float s = 0.f;
            for (int n = 0; n < KKB; ++n) s += rS[((locl * KKB + n) << 5) + c];
            rsumS[tid] = s;
        }
        __syncthreads();
        const float rsA = rsumS[(locA << 5) + cc];
        const float rsB = rsumS[(locB << 5) + cc];
        const float invA = __builtin_amdgcn_rcpf(rsA + EPSF);
        const float invB = __builtin_amdgcn_rcpf(rsB + EPSF);

        // ---- pass B: weighted mean / variance over n (votes via WMMA) ----
        float s1[8], s2[8];
#pragma unroll
        for (int g = 0; g < 8; ++g) { s1[g] = 0.f; s2[g] = 0.f; }
        for (int n = 0; n < KKB; ++n) {
            __builtin_prefetch(Wg + (n + 8) * CO * PS, 0, 1);
            v8f v = vote_tile(Wg, poseS, n, lane, wv);
            const float cA = rS[((locA * KKB + n) << 5) + cc] * invA;
            const float cB = rS[((locB * KKB + n) << 5) + cc] * invB;
#pragma unroll
            for (int g = 0; g < 4; ++g) { s1[g] += cA * v[g]; s2[g] += cA * v[g] * v[g]; }
#pragma unroll
            for (int g = 4; g < 8; ++g) { s1[g] += cB * v[g]; s2[g] += cB * v[g] * v[g]; }
        }
        float chA = 0.f, chB = 0.f;
#pragma unroll
        for (int g = 0; g < 8; ++g) {
            mu[g] = s1[g];
            sg[g] = s2[g] - mu[g] * mu[g];
            const float t = bu + 0.5f * logf(sg[g]);
            if (g < 4) chA += t; else chB += t;
        }
        chA = qsum4(chA) * rsA;
        chB = qsum4(chB) * rsB;
        const float aoA = __builtin_amdgcn_rcpf(1.f + expf(-LAM * (ba - chA)));
        const float aoB = __builtin_amdgcn_rcpf(1.f + expf(-LAM * (ba - chB)));
        logaA = logf(aoA);
        logaB = logf(aoB);

        if (it == 2) {
            // ---- write outputs: mu (b,h,w,C,16) then a_out (b,h,w,C,1) ----
#pragma unroll
            for (int g = 0; g < 8; ++g) {
                const int locl = (g < 4 ? locA : locB);
                const int i = g & 3;
                out[(locBase + locl) * (CO * PS) + cc * PS + i * 4 + l] = mu[g];
            }
            if (l == 0) {
                out[NLOC * CO * PS + (locBase + locA) * CO + cc] = aoA;
                out[NLOC * CO * PS + (locBase + locB) * CO + cc] = aoB;
            }
        }
    }
}

extern "C" void kernel_launch(void* const* d_in, const int* in_sizes, int n_in,
                              void* d_out, int out_size, void* d_ws, size_t ws_size,
                              hipStream_t stream) {
    (void)in_sizes; (void)n_in; (void)out_size; (void)d_ws; (void)ws_size;
    const float* x  = (const float*)d_in[0];
    const float* a  = (const float*)d_in[1];
    const float* w  = (const float*)d_in[2];
    const float* bu = (const float*)d_in[3];
    const float* ba = (const float*)d_in[4];
    float* out = (float*)d_out;
    const size_t shbytes = (size_t)SMEMF * sizeof(float);
    convcaps_em_kernel<<<dim3(NLOC / LPB), dim3(256), shbytes, stream>>>(
        x, a, w, bu, ba, out);
}